// MessagePassingLayer_22445499089015
// MI455X (gfx1250) — compile-verified
//
#include <hip/hip_runtime.h>
#include <math.h>

#define HD 128          // hidden dim = DIN = H = DOUT
#define NTILES (HD/16)  // 8

typedef float v2f __attribute__((ext_vector_type(2)));
typedef float v8f __attribute__((ext_vector_type(8)));

__device__ __forceinline__ float gelu_erf(float x) {
    return 0.5f * x * (1.0f + erff(x * 0.70710678118654752440f));
}
__device__ __forceinline__ float sigmoidf(float x) {
    return 1.0f / (1.0f + __expf(-x));
}

// Accumulate [16,HD](LDS, row-major) @ W[HD,HD] for one 16-column tile.
// WT is the TRANSPOSED weight (WT[n][k] = W[k][n]) so both A and B fragments
// are single contiguous 8-byte loads.
// A frag: lane m = lane&15, K pair {0,1} lanes 0-15, {2,3} lanes 16-31.
// B frag: lane n = lane&15, same K-pair split.
__device__ __forceinline__ v8f gemm_tile(const float* __restrict__ A,
                                         const float* __restrict__ WT,
                                         int ntile, int lane, v8f acc) {
    const int mn   = lane & 15;
    const int koff = (lane >> 4) * 2;
    const float* ap = A + mn * HD + koff;
    const float* bp = WT + (ntile * 16 + mn) * HD + koff;
    for (int k0 = 0; k0 < HD; k0 += 4) {
        v2f a = *reinterpret_cast<const v2f*>(ap + k0);   // A[m][k .. k+1]
        v2f b = *reinterpret_cast<const v2f*>(bp + k0);   // W[k][n], W[k+1][n]
        acc = __builtin_amdgcn_wmma_f32_16x16x4_f32(false, a, false, b,
                                                    (short)0, acc, false, false);
    }
    return acc;
}

__device__ __forceinline__ v8f bias_acc(const float* __restrict__ b, int ntile, int lane) {
    float bv = b[ntile * 16 + (lane & 15)];
    v8f acc = {bv, bv, bv, bv, bv, bv, bv, bv};
    return acc;
}

// Store D tile to LDS [16,HD]; m = r + 8*(lane>>4), n = lane&15.
template <bool GELU>
__device__ __forceinline__ void store_tile_lds(float* __restrict__ D, int ntile,
                                               int lane, v8f acc) {
    const int n  = lane & 15;
    const int hi = lane >> 4;
#pragma unroll
    for (int r = 0; r < 8; ++r) {
        float v = acc[r];
        if (GELU) v = gelu_erf(v);
        D[(r + hi * 8) * HD + ntile * 16 + n] = v;
    }
}

__global__ void zero_f32(float* __restrict__ p, size_t n) {
    size_t i = (size_t)blockIdx.x * blockDim.x + threadIdx.x;
    size_t stride = (size_t)gridDim.x * blockDim.x;
    for (; i < n; i += stride) p[i] = 0.0f;
}

// WT[n][k] = W[k][n]; HD x HD; grid 64 x 256 threads; coalesced writes.
__global__ __launch_bounds__(256) void transpose128(const float* __restrict__ W,
                                                    float* __restrict__ WT) {
    int idx = blockIdx.x * 256 + threadIdx.x;   // 0..HD*HD-1
    int n = idx >> 7, k = idx & (HD - 1);
    WT[n * HD + k] = W[k * HD + n];
}

// ---------------- node transform: src_h / dst_h = feat @ W + b ----------------
__global__ __launch_bounds__(256) void node_xform(
    const float* __restrict__ feat,
    const float* __restrict__ WsrcT, const float* __restrict__ bsrc,
    const float* __restrict__ WdstT, const float* __restrict__ bdst,
    float* __restrict__ src_h, float* __restrict__ dst_h, int n_nodes) {
    __shared__ float A[16 * HD];
    const int t = threadIdx.x;
    const int row0 = blockIdx.x * 16;

    {   // stage 16 feat rows (each thread: 32B = 8 floats)
        int r = t >> 4, seg = t & 15;
        int g = row0 + r; if (g >= n_nodes) g = n_nodes - 1;
        const float4* sp = reinterpret_cast<const float4*>(feat + (size_t)g * HD + seg * 8);
        float4 p0 = sp[0], p1 = sp[1];
        float4* dp = reinterpret_cast<float4*>(&A[r * HD + seg * 8]);
        dp[0] = p0; dp[1] = p1;
    }
    __syncthreads();

    const int wave = t >> 5, lane = t & 31;
    const int n = lane & 15, hi = lane >> 4;
    for (int task = wave; task < 16; task += 8) {
        const int which = task >> 3;
        const int ntile = task & 7;
        const float* WT = which ? WdstT : WsrcT;
        const float* b  = which ? bdst : bsrc;
        float* out      = which ? dst_h : src_h;
        v8f acc = bias_acc(b, ntile, lane);
        acc = gemm_tile(A, WT, ntile, lane, acc);
#pragma unroll
        for (int r = 0; r < 8; ++r) {
            int gm = row0 + r + hi * 8;
            if (gm < n_nodes) out[(size_t)gm * HD + ntile * 16 + n] = acc[r];
        }
    }
}

// ---------------- fused edge MLP + gate + LN + scatter ----------------
__global__ __launch_bounds__(256) void edge_mlp(
    const float* __restrict__ src_h, const float* __restrict__ dst_h,
    const int* __restrict__ esrc, const int* __restrict__ edst,
    const float* __restrict__ W1aT, const float* __restrict__ W1bT, const float* __restrict__ b1,
    const float* __restrict__ W2T,  const float* __restrict__ b2,
    const float* __restrict__ W3T,  const float* __restrict__ b3,
    const float* __restrict__ ln_g, const float* __restrict__ ln_b,
    const float* __restrict__ Wg1aT, const float* __restrict__ Wg1bT, const float* __restrict__ bg1,
    const float* __restrict__ Wg2,   const float* __restrict__ bg2,
    float* __restrict__ upd, int n_edges) {
    __shared__ float SE[16 * HD];   // se  -> later h2
    __shared__ float DE[16 * HD];   // de  -> later msg (pre-LN)
    __shared__ float H1[16 * HD];   // gelu(se@W1a + de@W1b + b1)
    __shared__ float G [16 * HD];   // gelu(se@Wg1a + de@Wg1b + bg1)
    __shared__ int   s_src[16], s_dst[16];
    __shared__ float s_mean[16], s_rstd[16], s_gate[16];

    const int t  = threadIdx.x;
    const int e0 = blockIdx.x * 16;
    if (t < 16) {
        int e = e0 + t; if (e >= n_edges) e = n_edges - 1;
        s_src[t] = esrc[e];
        s_dst[t] = edst[e];
    }
    __syncthreads();

    const int r = t >> 4, seg = t & 15;
    {   // gather se/de rows
        const float4* sp = reinterpret_cast<const float4*>(src_h + (size_t)s_src[r] * HD + seg * 8);
        const float4* dp = reinterpret_cast<const float4*>(dst_h + (size_t)s_dst[r] * HD + seg * 8);
        float4 a0 = sp[0], a1 = sp[1], c0 = dp[0], c1 = dp[1];
        float4* so = reinterpret_cast<float4*>(&SE[r * HD + seg * 8]);
        float4* dd = reinterpret_cast<float4*>(&DE[r * HD + seg * 8]);
        so[0] = a0; so[1] = a1; dd[0] = c0; dd[1] = c1;
    }
    __syncthreads();

    const int wave = t >> 5, lane = t & 31;   // 8 waves == 8 n-tiles
    const int ntile = wave;

    // h1 = gelu(se@W1a + de@W1b + b1)
    v8f acc = bias_acc(b1, ntile, lane);
    acc = gemm_tile(SE, W1aT, ntile, lane, acc);
    acc = gemm_tile(DE, W1bT, ntile, lane, acc);
    store_tile_lds<true>(H1, ntile, lane, acc);
    // g = gelu(se@Wg1a + de@Wg1b + bg1)
    acc = bias_acc(bg1, ntile, lane);
    acc = gemm_tile(SE, Wg1aT, ntile, lane, acc);
    acc = gemm_tile(DE, Wg1bT, ntile, lane, acc);
    store_tile_lds<true>(G, ntile, lane, acc);
    __syncthreads();

    // h2 = gelu(h1@W2 + b2) -> SE
    acc = bias_acc(b2, ntile, lane);
    acc = gemm_tile(H1, W2T, ntile, lane, acc);
    store_tile_lds<true>(SE, ntile, lane, acc);
    __syncthreads();

    // msg_pre = h2@W3 + b3 -> DE
    acc = bias_acc(b3, ntile, lane);
    acc = gemm_tile(SE, W3T, ntile, lane, acc);
    store_tile_lds<false>(DE, ntile, lane, acc);
    __syncthreads();

    // per-row LN stats (lanes 0-15) and gate GEMV (lanes 16-31)
    if (t < 16) {
        const float* row = &DE[t * HD];
        float s = 0.f, s2 = 0.f;
        for (int c = 0; c < HD; ++c) { float x = row[c]; s += x; s2 += x * x; }
        float mu = s * (1.0f / HD);
        float var = s2 * (1.0f / HD) - mu * mu;
        s_mean[t] = mu;
        s_rstd[t] = rsqrtf(var + 1e-5f);
    } else if (t < 32) {
        const int i = t - 16;
        const float* row = &G[i * HD];
        float d = 0.f;
        for (int c = 0; c < HD; ++c) d += row[c] * Wg2[c];
        s_gate[i] = sigmoidf(d + bg2[0]);
    }
    __syncthreads();

    // gated scatter-add: upd[dst] += gate * layernorm(msg)
    if (e0 + r < n_edges) {
        const float gate = s_gate[r], mu = s_mean[r], rs = s_rstd[r];
        float* base = upd + (size_t)s_dst[r] * HD + seg * 8;
#pragma unroll
        for (int j = 0; j < 8; ++j) {
            int c = seg * 8 + j;
            float x = (DE[r * HD + c] - mu) * rs * ln_g[c] + ln_b[c];
            atomicAdd(base + j, gate * x);
        }
    }
}

// ---------------- output transform: out = upd @ W_out + b_out ----------------
__global__ __launch_bounds__(256) void out_xform(
    const float* __restrict__ upd,
    const float* __restrict__ WoutT, const float* __restrict__ bout,
    float* __restrict__ out, int n_nodes) {
    __shared__ float A[16 * HD];
    const int t = threadIdx.x;
    const int row0 = blockIdx.x * 16;
    {
        int r = t >> 4, seg = t & 15;
        int g = row0 + r; if (g >= n_nodes) g = n_nodes - 1;
        const float4* sp = reinterpret_cast<const float4*>(upd + (size_t)g * HD + seg * 8);
        float4 p0 = sp[0], p1 = sp[1];
        float4* dp = reinterpret_cast<float4*>(&A[r * HD + seg * 8]);
        dp[0] = p0; dp[1] = p1;
    }
    __syncthreads();
    const int wave = t >> 5, lane = t & 31;
    const int ntile = wave;
    const int n = lane & 15, hi = lane >> 4;
    v8f acc = bias_acc(bout, ntile, lane);
    acc = gemm_tile(A, WoutT, ntile, lane, acc);
#pragma unroll
    for (int r = 0; r < 8; ++r) {
        int gm = row0 + r + hi * 8;
        if (gm < n_nodes) out[(size_t)gm * HD + ntile * 16 + n] = acc[r];
    }
}

extern "C" void kernel_launch(void* const* d_in, const int* in_sizes, int n_in,
                              void* d_out, int out_size, void* d_ws, size_t ws_size,
                              hipStream_t stream) {
    const float* feat = (const float*)d_in[0];
    const int*   esrc = (const int*)d_in[1];
    const int*   edst = (const int*)d_in[2];
    const float* W_src = (const float*)d_in[3];
    const float* b_src = (const float*)d_in[4];
    const float* W_dst = (const float*)d_in[5];
    const float* b_dst = (const float*)d_in[6];
    const float* W1a = (const float*)d_in[7];
    const float* W1b = (const float*)d_in[8];
    const float* b1  = (const float*)d_in[9];
    const float* W2  = (const float*)d_in[10];
    const float* b2  = (const float*)d_in[11];
    const float* W3  = (const float*)d_in[12];
    const float* b3  = (const float*)d_in[13];
    const float* ln_g = (const float*)d_in[14];
    const float* ln_b = (const float*)d_in[15];
    const float* Wg1a = (const float*)d_in[16];
    const float* Wg1b = (const float*)d_in[17];
    const float* bg1  = (const float*)d_in[18];
    const float* Wg2  = (const float*)d_in[19];
    const float* bg2  = (const float*)d_in[20];
    const float* W_out = (const float*)d_in[21];
    const float* b_out = (const float*)d_in[22];

    const int n_nodes = in_sizes[0] / HD;
    const int n_edges = in_sizes[1];

    float* ws    = (float*)d_ws;
    float* upd   = ws;                              // [N,HD]
    float* src_h = ws + (size_t)n_nodes * HD;       // [N,HD]
    float* dst_h = ws + 2 * (size_t)n_nodes * HD;   // [N,HD]
    float* wt    = ws + 3 * (size_t)n_nodes * HD;   // 9 transposed HDxHD weights
    float* W_srcT = wt + 0 * HD * HD;
    float* W_dstT = wt + 1 * HD * HD;
    float* W1aT   = wt + 2 * HD * HD;
    float* W1bT   = wt + 3 * HD * HD;
    float* W2T    = wt + 4 * HD * HD;
    float* W3T    = wt + 5 * HD * HD;
    float* Wg1aT  = wt + 6 * HD * HD;
    float* Wg1bT  = wt + 7 * HD * HD;
    float* W_outT = wt + 8 * HD * HD;

    const int tgrid = (HD * HD) / 256;  // 64 blocks
    transpose128<<<tgrid, 256, 0, stream>>>(W_src, W_srcT);
    transpose128<<<tgrid, 256, 0, stream>>>(W_dst, W_dstT);
    transpose128<<<tgrid, 256, 0, stream>>>(W1a,   W1aT);
    transpose128<<<tgrid, 256, 0, stream>>>(W1b,   W1bT);
    transpose128<<<tgrid, 256, 0, stream>>>(W2,    W2T);
    transpose128<<<tgrid, 256, 0, stream>>>(W3,    W3T);
    transpose128<<<tgrid, 256, 0, stream>>>(Wg1a,  Wg1aT);
    transpose128<<<tgrid, 256, 0, stream>>>(Wg1b,  Wg1bT);
    transpose128<<<tgrid, 256, 0, stream>>>(W_out, W_outT);

    zero_f32<<<4096, 256, 0, stream>>>(upd, (size_t)n_nodes * HD);
    node_xform<<<(n_nodes + 15) / 16, 256, 0, stream>>>(
        feat, W_srcT, b_src, W_dstT, b_dst, src_h, dst_h, n_nodes);
    edge_mlp<<<(n_edges + 15) / 16, 256, 0, stream>>>(
        src_h, dst_h, esrc, edst, W1aT, W1bT, b1, W2T, b2, W3T, b3,
        ln_g, ln_b, Wg1aT, Wg1bT, bg1, Wg2, bg2, upd, n_edges);
    out_xform<<<(n_nodes + 15) / 16, 256, 0, stream>>>(
        upd, W_outT, b_out, (float*)d_out, n_nodes);
}